// GEMAL_20615843021206
// MI455X (gfx1250) — compile-verified
//
#include <hip/hip_runtime.h>
#include <hip/hip_bf16.h>
#include <math.h>

#define N_NODES   50000
#define N_EDGES   800000
#define N_FEAT    128
#define HIDDEN    64
#define EMBED     300
#define N_CLASSES 10
#define N_GRAPHS  500
#define LEAKY     0.01f

typedef __attribute__((ext_vector_type(2))) float v2f;
typedef __attribute__((ext_vector_type(8))) float v8f;

static __device__ __forceinline__ void atomicMaxF(float* addr, float val) {
    int* ai = (int*)addr;
    int old = __float_as_int(*addr);
    while (__int_as_float(old) < val) {
        int assumed = old;
        old = atomicCAS(ai, assumed, __float_as_int(val));
        if (old == assumed) break;
    }
}

// ---------------- utility ----------------
__global__ void fill_kernel(float* __restrict__ p, float v, int n) {
    int i = blockIdx.x * blockDim.x + threadIdx.x;
    if (i < n) p[i] = v;
}

// ---------------- degree / norm ----------------
__global__ void deg_kernel(const long long* __restrict__ dst, float* __restrict__ deg, int n_edges) {
    int e = blockIdx.x * blockDim.x + threadIdx.x;
    if (e < n_edges) atomicAdd(&deg[dst[e]], 1.0f);
}

__global__ void dis_kernel(float* __restrict__ deg, int n) {
    int i = blockIdx.x * blockDim.x + threadIdx.x;
    if (i < n) deg[i] = rsqrtf(deg[i] + 1.0f);   // in-place: deg -> deg^{-1/2}
}

// ---------------- WMMA f32 GEMM: C[M,N] = A[M,K] @ B[K,N], K,N compile-time ----------------
// One wave computes a full 16 x N row-strip (N/16 accumulators); A loaded once per K-step
// and shared across all N-tiles. All B/C offsets are compile-time immediates.
// f32 16x16x4 layouts (ISA 7.12.2):
//   A 16x4 : lanes 0-15 = M=0..15 hold {K=0,K=1}; lanes 16-31 hold {K=2,K=3}
//   B 4x16 : mirror (lanes = N, half-wave = K pairs)
//   C 16x16: vgpr r = row (half*8 + r), lane&15 = col
template <int K, int N>
__global__ void wmma_gemm_f32(const float* __restrict__ A, const float* __restrict__ B,
                              float* __restrict__ C) {
    constexpr int NT = N / 16;
    const int mtile = blockIdx.x;
    const int lane  = threadIdx.x;      // 0..31
    const int half  = lane >> 4;        // 0 or 1
    const int l15   = lane & 15;

    const float* Arow  = A + (size_t)(mtile * 16 + l15) * K + 2 * half;  // 8B aligned
    const float* Bbase = B + 2 * half * N + l15;

    v8f acc[NT];
#pragma unroll
    for (int t = 0; t < NT; ++t) acc[t] = (v8f){};

#pragma unroll
    for (int k = 0; k < K; k += 4) {
        v2f a = *(const v2f*)(Arow + k);          // one global_load_b64
#pragma unroll
        for (int t = 0; t < NT; ++t) {
            v2f b;
            b.x = Bbase[k * N + t * 16];          // immediate offsets
            b.y = Bbase[k * N + N + t * 16];
            acc[t] = __builtin_amdgcn_wmma_f32_16x16x4_f32(false, a, false, b,
                                                           (short)0, acc[t], false, false);
        }
    }

    float* Crow = C + (size_t)(mtile * 16 + half * 8) * N + l15;
#pragma unroll
    for (int t = 0; t < NT; ++t)
#pragma unroll
        for (int r = 0; r < 8; ++r) Crow[r * N + t * 16] = acc[t][r];
}

// ---------------- edge scatter: agg[dst] += hw[src] * dis[src]*dis[dst] ----------------
// 16 threads per edge, 4 floats per thread (float4 gather, f32 global atomics).
__global__ void scatter_kernel(const long long* __restrict__ ei, const float* __restrict__ hw,
                               const float* __restrict__ dis, float* __restrict__ agg) {
    int tid = blockIdx.x * blockDim.x + threadIdx.x;
    int e = tid >> 4;
    if (e >= N_EDGES) return;
    int c = (tid & 15) * 4;
    long long s = ei[e];
    long long d = ei[N_EDGES + e];
    float norm = dis[s] * dis[d];
    float4 v = *(const float4*)(hw + (size_t)s * HIDDEN + c);
    float* ap = agg + (size_t)d * HIDDEN + c;
    atomicAdd(ap + 0, v.x * norm);
    atomicAdd(ap + 1, v.y * norm);
    atomicAdd(ap + 2, v.z * norm);
    atomicAdd(ap + 3, v.w * norm);
}

// ---------------- h = relu(agg + hw * dis^2 + b) ----------------
__global__ void finalize_kernel(const float* __restrict__ agg, const float* __restrict__ hw,
                                const float* __restrict__ dis, const float* __restrict__ b,
                                float* __restrict__ out, int total) {
    int i = blockIdx.x * blockDim.x + threadIdx.x;
    if (i >= total) return;
    int n = i >> 6;      // /HIDDEN
    int c = i & 63;
    float d = dis[n];
    float v = agg[i] + hw[i] * d * d + b[c];
    out[i] = v > 0.f ? v : 0.f;
}

// ---------------- attention score ----------------
__global__ void att_kernel(const float* __restrict__ h, const float* __restrict__ att_w,
                           const float* __restrict__ att_b, float* __restrict__ s) {
    int n = blockIdx.x * blockDim.x + threadIdx.x;
    if (n >= N_NODES) return;
    const float* hp = h + (size_t)n * HIDDEN;
    float acc = att_b[0];
#pragma unroll
    for (int c = 0; c < HIDDEN; ++c) acc += hp[c] * att_w[c];
    s[n] = acc > 0.f ? acc : LEAKY * acc;
}

__global__ void segmax_kernel(const float* __restrict__ s, const long long* __restrict__ batch,
                              float* __restrict__ m) {
    int i = blockIdx.x * blockDim.x + threadIdx.x;
    if (i < N_NODES) atomicMaxF(&m[batch[i]], s[i]);
}

__global__ void expsum_kernel(float* __restrict__ s, const long long* __restrict__ batch,
                              const float* __restrict__ m, float* __restrict__ z) {
    int i = blockIdx.x * blockDim.x + threadIdx.x;
    if (i >= N_NODES) return;
    long long g = batch[i];
    float e = expf(s[i] - m[g]);
    s[i] = e;                       // overwrite score with exp term
    atomicAdd(&z[g], e);
}

__global__ void pool_kernel(const float* __restrict__ h, const float* __restrict__ e,
                            const float* __restrict__ z, const long long* __restrict__ batch,
                            float* __restrict__ gfeat, int total) {
    int i = blockIdx.x * blockDim.x + threadIdx.x;
    if (i >= total) return;
    int n = i >> 6;
    int c = i & 63;
    long long g = batch[n];
    float coef = e[n] / z[g];
    atomicAdd(&gfeat[(size_t)g * HIDDEN + c], coef * h[i]);
}

// ---------------- small dense layers: out[M,N] = act(A[M,K] @ W[K,N] + b) ----------------
__global__ void dense_kernel(const float* __restrict__ A, const float* __restrict__ W,
                             const float* __restrict__ b, float* __restrict__ out,
                             int M, int K, int N, int do_relu) {
    int i = blockIdx.x * blockDim.x + threadIdx.x;
    if (i >= M * N) return;
    int m = i / N, n = i % N;
    const float* a = A + (size_t)m * K;
    const float* w = W + n;
    float acc = b[n];
    for (int k = 0; k < K; ++k) acc += a[k] * w[(size_t)k * N];
    if (do_relu && acc < 0.f) acc = 0.f;
    out[i] = acc;
}

static inline int cdiv(long long a, int b) { return (int)((a + b - 1) / b); }

extern "C" void kernel_launch(void* const* d_in, const int* in_sizes, int n_in,
                              void* d_out, int out_size, void* d_ws, size_t ws_size,
                              hipStream_t stream) {
    const float*     x      = (const float*)d_in[0];
    const long long* ei     = (const long long*)d_in[1];   // [2, E] int64
    const long long* batch  = (const long long*)d_in[2];   // [N] int64
    const float*     W1     = (const float*)d_in[3];
    const float*     b1     = (const float*)d_in[4];
    const float*     W2     = (const float*)d_in[5];
    const float*     b2     = (const float*)d_in[6];
    const float*     att_w  = (const float*)d_in[7];
    const float*     att_b  = (const float*)d_in[8];
    const float*     proj_w = (const float*)d_in[9];
    const float*     proj_b = (const float*)d_in[10];
    const float*     cls_w1 = (const float*)d_in[11];
    const float*     cls_b1 = (const float*)d_in[12];
    const float*     cls_w2 = (const float*)d_in[13];
    const float*     cls_b2 = (const float*)d_in[14];
    float* out = (float*)d_out;

    // workspace bump allocator
    char* wsp = (char*)d_ws;
    auto alloc = [&](size_t bytes) -> float* {
        float* p = (float*)wsp;
        wsp += (bytes + 255) & ~(size_t)255;
        return p;
    };
    float* hw    = alloc((size_t)N_NODES * HIDDEN * 4);   // node features after @W
    float* agg   = alloc((size_t)N_NODES * HIDDEN * 4);   // edge-aggregated
    float* h1    = alloc((size_t)N_NODES * HIDDEN * 4);
    float* h2    = alloc((size_t)N_NODES * HIDDEN * 4);
    float* dis   = alloc((size_t)N_NODES * 4);            // deg -> deg^{-1/2} in place
    float* satt  = alloc((size_t)N_NODES * 4);            // score -> exp term in place
    float* m     = alloc((size_t)N_GRAPHS * 4);
    float* z     = alloc((size_t)N_GRAPHS * 4);
    float* gfeat = alloc((size_t)N_GRAPHS * HIDDEN * 4);
    float* g300  = alloc((size_t)N_GRAPHS * EMBED * 4);
    float* g128  = alloc((size_t)N_GRAPHS * 128 * 4);

    const int B = 256;
    const int NH = N_NODES * HIDDEN;

    // degrees -> dis (shared by both convs)
    fill_kernel<<<cdiv(N_NODES, B), B, 0, stream>>>(dis, 0.0f, N_NODES);
    deg_kernel<<<cdiv(N_EDGES, B), B, 0, stream>>>(ei + N_EDGES, dis, N_EDGES);
    dis_kernel<<<cdiv(N_NODES, B), B, 0, stream>>>(dis, N_NODES);

    // ---- conv1: hw = x @ W1 ; aggregate ; h1 = relu(agg + hw*dis^2 + b1)
    wmma_gemm_f32<N_FEAT, HIDDEN><<<N_NODES / 16, dim3(32), 0, stream>>>(x, W1, hw);
    fill_kernel<<<cdiv(NH, B), B, 0, stream>>>(agg, 0.0f, NH);
    scatter_kernel<<<cdiv((long long)N_EDGES * 16, B), B, 0, stream>>>(ei, hw, dis, agg);
    finalize_kernel<<<cdiv(NH, B), B, 0, stream>>>(agg, hw, dis, b1, h1, NH);

    // ---- conv2: hw = h1 @ W2 ; aggregate ; h2 = relu(agg + hw*dis^2 + b2)
    wmma_gemm_f32<HIDDEN, HIDDEN><<<N_NODES / 16, dim3(32), 0, stream>>>(h1, W2, hw);
    fill_kernel<<<cdiv(NH, B), B, 0, stream>>>(agg, 0.0f, NH);
    scatter_kernel<<<cdiv((long long)N_EDGES * 16, B), B, 0, stream>>>(ei, hw, dis, agg);
    finalize_kernel<<<cdiv(NH, B), B, 0, stream>>>(agg, hw, dis, b2, h2, NH);

    // ---- attention + segment softmax + weighted pooling
    att_kernel<<<cdiv(N_NODES, B), B, 0, stream>>>(h2, att_w, att_b, satt);
    fill_kernel<<<cdiv(N_GRAPHS, B), B, 0, stream>>>(m, -INFINITY, N_GRAPHS);
    fill_kernel<<<cdiv(N_GRAPHS, B), B, 0, stream>>>(z, 0.0f, N_GRAPHS);
    fill_kernel<<<cdiv(N_GRAPHS * HIDDEN, B), B, 0, stream>>>(gfeat, 0.0f, N_GRAPHS * HIDDEN);
    segmax_kernel<<<cdiv(N_NODES, B), B, 0, stream>>>(satt, batch, m);
    expsum_kernel<<<cdiv(N_NODES, B), B, 0, stream>>>(satt, batch, m, z);
    pool_kernel<<<cdiv(NH, B), B, 0, stream>>>(h2, satt, z, batch, gfeat, NH);

    // ---- head: proj -> relu(cls1) -> cls2
    dense_kernel<<<cdiv(N_GRAPHS * EMBED, B), B, 0, stream>>>(gfeat, proj_w, proj_b, g300,
                                                              N_GRAPHS, HIDDEN, EMBED, 0);
    dense_kernel<<<cdiv(N_GRAPHS * 128, B), B, 0, stream>>>(g300, cls_w1, cls_b1, g128,
                                                            N_GRAPHS, EMBED, 128, 1);
    dense_kernel<<<cdiv(N_GRAPHS * N_CLASSES, B), B, 0, stream>>>(g128, cls_w2, cls_b2, out,
                                                                  N_GRAPHS, 128, N_CLASSES, 0);
}